// MessagePassing_42588895707817
// MI455X (gfx1250) — compile-verified
//
#include <hip/hip_runtime.h>

typedef float v2f __attribute__((ext_vector_type(2)));
typedef float v8f __attribute__((ext_vector_type(8)));

#define FDIM 128
#define GEPS 1e-10f

// ---------------------------------------------------------------------------
// prep: M[f][g] = sum_h W[h][f] * graph[h][g] * invdeg[h]
//       bias2[g] = sum_h b[h]   * graph[h][g] * invdeg[h]
// invdeg[h] = (rowsum(graph[h]) > EPS) ? 1/rowsum : 0   (matches jnp.where+clip)
// Grid: 128 blocks (one per f) x 128 threads (one per g). ~2M FMAs total.
// ---------------------------------------------------------------------------
__global__ __launch_bounds__(128) void prep_kernel(const float* __restrict__ graph,
                                                   const float* __restrict__ W,
                                                   const float* __restrict__ b,
                                                   float* __restrict__ M,
                                                   float* __restrict__ bias2) {
    __shared__ float invdeg[FDIM];
    const int g = threadIdx.x;
    const int f = blockIdx.x;

    float s = 0.0f;
    for (int c = 0; c < FDIM; ++c) s += graph[g * FDIM + c];
    invdeg[g] = (s > GEPS) ? (1.0f / s) : 0.0f;
    __syncthreads();

    float acc = 0.0f;
    for (int hh = 0; hh < FDIM; ++hh)
        acc = fmaf(W[hh * FDIM + f], graph[hh * FDIM + g] * invdeg[hh], acc);
    M[f * FDIM + g] = acc;

    if (f == 0) {
        float bb = 0.0f;
        for (int hh = 0; hh < FDIM; ++hh)
            bb = fmaf(b[hh], graph[hh * FDIM + g] * invdeg[hh], bb);
        bias2[g] = bb;
    }
}

// ---------------------------------------------------------------------------
// Main GEMM: out[r][g] = sum_f h[r][f] * M[f][g] + bias2[g]
//
// LDS layout: pair-interleaved. Pair-row p (= k/2) holds float2
//   {M[2p][c], M[2p+1][c]} at float2-index p*128 + (c ^ ((p&1)<<4)).
// => B fragment (K=kk,kk+1 at column c) is ONE aligned ds_load_b64 with a
//    compile-time immediate offset (ks*2048B), and the XOR-16 column swizzle
//    puts the two wave halves (pair-rows 2ks / 2ks+1) on complementary bank
//    sets -> conflict-free.
// ---------------------------------------------------------------------------
__global__ __launch_bounds__(256) void gemm_kernel(const float* __restrict__ hmat,
                                                   const float* __restrict__ M,
                                                   const float* __restrict__ bias2,
                                                   float* __restrict__ out) {
    __shared__ float ldsM[FDIM * FDIM];   // 64 KB, pair-interleaved + swizzled

    const int tid = threadIdx.x;
    for (int i = tid; i < FDIM * FDIM; i += 256) {
        const int r = i >> 7, c = i & 127;
        const int p = r >> 1, comp = r & 1;
        const int colAdj = c ^ ((p & 1) << 4);
        ldsM[p * 256 + colAdj * 2 + comp] = M[i];
    }
    __syncthreads();

    const int wave  = tid >> 5;           // 0..7
    const int lane  = tid & 31;
    const int laneM = lane & 15;          // M-row / N-col within tile
    const int half  = lane >> 4;          // 0 or 1
    const int kOff  = half << 1;          // 0 (lanes 0-15) or 2 (lanes 16-31)

    const long rowBase = (long)blockIdx.x * 128 + (long)wave * 16;
    const float* hrow = hmat + (rowBase + laneM) * FDIM;

    // Preload all A fragments for this wave's 16 rows: 16x128 f32 -> 64 VGPRs.
    // ISA A-layout (16x4 f32): lane<16 holds (K=k,K=k+1), lane>=16 holds (K=k+2,K=k+3).
    v2f a[32];
#pragma unroll
    for (int ks = 0; ks < 32; ++ks)
        a[ks] = *(const v2f*)(hrow + ks * 4 + kOff);

    for (int n = 0; n < 8; ++n) {
        const int col = n * 16 + laneM;
        const float bsc = bias2[col];
        v8f acc = { bsc, bsc, bsc, bsc, bsc, bsc, bsc, bsc };

        // Per-lane base: pair-row (2*0 + half), column (col ^ (half*16)).
        const v2f* bptr = (const v2f*)&ldsM[half * 256 + ((col ^ (half << 4)) << 1)];

#pragma unroll
        for (int ks = 0; ks < 32; ++ks) {
            const v2f bfrag = bptr[ks * 256];   // pair-rows 2ks(+half): imm offset ks*2048B
            acc = __builtin_amdgcn_wmma_f32_16x16x4_f32(
                false, a[ks], false, bfrag, (short)0, acc, false, false);
        }

        // C/D layout: VGPR r -> (row r, lanes 0-15) / (row r+8, lanes 16-31), N = lane%16.
        float* o = out + (rowBase + (long)(half << 3)) * FDIM + n * 16 + laneM;
#pragma unroll
        for (int r = 0; r < 8; ++r)
            o[(long)r * FDIM] = acc[r];
    }
}

extern "C" void kernel_launch(void* const* d_in, const int* in_sizes, int n_in,
                              void* d_out, int out_size, void* d_ws, size_t ws_size,
                              hipStream_t stream) {
    const float* h     = (const float*)d_in[0];   // [B,T,128] f32
    const float* graph = (const float*)d_in[1];   // [128,128] f32
    const float* W     = (const float*)d_in[2];   // [128,128] f32
    const float* b     = (const float*)d_in[3];   // [128]     f32
    float* out = (float*)d_out;

    float* M     = (float*)d_ws;          // 128*128 f32 = 64 KB
    float* bias2 = M + FDIM * FDIM;       // 128 f32

    prep_kernel<<<dim3(FDIM), dim3(FDIM), 0, stream>>>(graph, W, b, M, bias2);

    const long rows = (long)out_size / FDIM;      // 262144
    const int nblocks = (int)(rows / 128);        // 2048
    gemm_kernel<<<dim3(nblocks), dim3(256), 0, stream>>>(h, M, bias2, out);
}